// TopKTiledSoftmax_14267881357589
// MI455X (gfx1250) — compile-verified
//
#include <hip/hip_runtime.h>
#include <math.h>

// ---------------------------------------------------------------------------
// Problem constants (fixed by the reference: setup_inputs()).
// ---------------------------------------------------------------------------
#define T_TOK  1024
#define DDIM   512
#define VOCAB  50304
#define KSEL   128
#define TILEF  16
#define TILE   (VOCAB / TILEF)      // 3144 columns per vocab tile
#define CAND   (TILEF * KSEL)       // 2048 candidates per token
#define ROWPAD 132                  // padded per-row sparse storage (>=129)
#define MROWS  64                   // token rows staged in LDS per block

typedef __attribute__((ext_vector_type(2))) float v2f;
typedef __attribute__((ext_vector_type(8))) float v8f;

// ---------------------------------------------------------------------------
// Kernel 1: fp32 GEMM  logits[t, v] = sum_d x[t,d] * W[v,d]
//   - 64 x-rows staged in LDS (128 KB, CDNA5 WGP allows up to 320 KB)
//   - each wave: one 16-col tile of W, FOUR 16-row accumulators -> each
//     global_load_b64 of B feeds 4x v_wmma_f32_16x16x4_f32 (4x B reuse,
//     cuts W L2 traffic from 6.6 GB to 1.65 GB for this 52.8 GFLOP GEMM)
//   A layout (16x4 f32): lanes 0-15 M=0..15; VGPR0 holds K=0 (lo half) /
//   K=2 (hi half), VGPR1 holds K=1 / K=3.  B layout mirrored for 4x16.
// ---------------------------------------------------------------------------
__global__ __launch_bounds__(128) void gemm_logits(const float* __restrict__ x,
                                                   const float* __restrict__ w,
                                                   float* __restrict__ out) {
    __shared__ float lds_x[MROWS * DDIM];         // 128 KB
    const int rb  = blockIdx.y;                   // 64-row block
    const int cb  = blockIdx.x;                   // 64-col block (4 waves x 16)
    const int tid = threadIdx.x;

    // Cooperative load of the x tile (64 rows x 512 floats), float4 vectors.
    {
        const float4* xsrc = (const float4*)(x + (size_t)rb * MROWS * DDIM);
        float4* xdst = (float4*)lds_x;
        #pragma unroll 8
        for (int i = tid; i < MROWS * DDIM / 4; i += 128) xdst[i] = xsrc[i];
    }
    __syncthreads();

    const int wave = tid >> 5;
    const int lane = tid & 31;
    const int half = lane >> 4;                   // 0: lanes 0-15, 1: lanes 16-31
    const int l16  = lane & 15;
    const int koff = half * 2;                    // K sub-offset per half

    const int c0 = (cb * 4 + wave) * 16;          // first vocab column of tile
    const float* wrow = w + (size_t)(c0 + l16) * DDIM;   // B source (row of W)
    const float* a0 = lds_x + (0 * 16 + l16) * DDIM;     // A sources (LDS)
    const float* a1 = lds_x + (1 * 16 + l16) * DDIM;
    const float* a2 = lds_x + (2 * 16 + l16) * DDIM;
    const float* a3 = lds_x + (3 * 16 + l16) * DDIM;

    v8f acc0 = {}, acc1 = {}, acc2 = {}, acc3 = {};
    #pragma unroll 4
    for (int k = 0; k < DDIM; k += 4) {
        if ((k & 31) == 0) __builtin_prefetch(wrow + k + 32, 0, 3);
        v2f b  = *(const v2f*)(wrow + k + koff);  // global_load_b64 (L2 hot)
        v2f q0 = *(const v2f*)(a0 + k + koff);    // ds_load_b64
        v2f q1 = *(const v2f*)(a1 + k + koff);
        v2f q2 = *(const v2f*)(a2 + k + koff);
        v2f q3 = *(const v2f*)(a3 + k + koff);
        acc0 = __builtin_amdgcn_wmma_f32_16x16x4_f32(false, q0, false, b,
                                                     (short)0, acc0, false, false);
        acc1 = __builtin_amdgcn_wmma_f32_16x16x4_f32(false, q1, false, b,
                                                     (short)0, acc1, false, false);
        acc2 = __builtin_amdgcn_wmma_f32_16x16x4_f32(false, q2, false, b,
                                                     (short)0, acc2, false, false);
        acc3 = __builtin_amdgcn_wmma_f32_16x16x4_f32(false, q3, false, b,
                                                     (short)0, acc3, false, false);
    }

    // C/D layout: lane<16 -> N=lane, M=g; lane>=16 -> N=lane-16, M=g+8.
    const int rbase = rb * MROWS + half * 8;
    float* dst0 = out + (size_t)(rbase +  0) * VOCAB + c0 + l16;
    float* dst1 = out + (size_t)(rbase + 16) * VOCAB + c0 + l16;
    float* dst2 = out + (size_t)(rbase + 32) * VOCAB + c0 + l16;
    float* dst3 = out + (size_t)(rbase + 48) * VOCAB + c0 + l16;
    #pragma unroll
    for (int g = 0; g < 8; ++g) {
        dst0[(size_t)g * VOCAB] = acc0[g];
        dst1[(size_t)g * VOCAB] = acc1[g];
        dst2[(size_t)g * VOCAB] = acc2[g];
        dst3[(size_t)g * VOCAB] = acc3[g];
    }
}

// ---------------------------------------------------------------------------
// Kernel 2: pull target logit per token (before d_out is recycled).
// ---------------------------------------------------------------------------
__global__ void grab_target(const float* __restrict__ logits,
                            const int* __restrict__ target,
                            float* __restrict__ tgt_val) {
    int t = blockIdx.x * blockDim.x + threadIdx.x;
    if (t < T_TOK) tgt_val[t] = logits[(size_t)t * VOCAB + target[t]];
}

// Monotonic float <-> uint mapping (order-preserving for radix select).
__device__ __forceinline__ unsigned f2key(float f) {
    unsigned u = __float_as_uint(f);
    return (u & 0x80000000u) ? ~u : (u | 0x80000000u);
}
__device__ __forceinline__ float key2f(unsigned k) {
    unsigned u = (k & 0x80000000u) ? (k & 0x7fffffffu) : ~k;
    return __uint_as_float(u);
}

// ---------------------------------------------------------------------------
// Kernel 3: per (token, tile) top-128 of 3144 via 32-pass MSB radix select.
// ---------------------------------------------------------------------------
__global__ __launch_bounds__(256) void tile_topk(const float* __restrict__ logits,
                                                 float* __restrict__ cand_val,
                                                 int* __restrict__ cand_idx) {
    __shared__ unsigned keys[TILE];               // 12.6 KB
    __shared__ int s_cnt, s_hi, s_eq;
    const int t    = blockIdx.y;
    const int tile = blockIdx.x;
    const int tid  = threadIdx.x;

    const float* row = logits + (size_t)t * VOCAB + (size_t)tile * TILE;
    for (int i = tid; i < TILE; i += 256) keys[i] = f2key(row[i]);
    __syncthreads();

    // Binary search for tau = 128th largest key.
    unsigned pivot = 0;
    for (int bit = 31; bit >= 0; --bit) {
        unsigned trial = pivot | (1u << bit);
        if (tid == 0) s_cnt = 0;
        __syncthreads();
        int c = 0;
        for (int i = tid; i < TILE; i += 256) c += (keys[i] >= trial);
        atomicAdd(&s_cnt, c);
        __syncthreads();
        if (s_cnt >= KSEL) pivot = trial;
        __syncthreads();
    }
    const unsigned tau = pivot;

    if (tid == 0) s_hi = 0;
    __syncthreads();
    int cg = 0;
    for (int i = tid; i < TILE; i += 256) cg += (keys[i] > tau);
    atomicAdd(&s_hi, cg);
    __syncthreads();
    const int n_hi = s_hi;
    if (tid == 0) { s_hi = 0; s_eq = 0; }
    __syncthreads();

    float* cv = cand_val + (size_t)t * CAND + tile * KSEL;
    int*   ci = cand_idx + (size_t)t * CAND + tile * KSEL;
    for (int i = tid; i < TILE; i += 256) {
        unsigned kk = keys[i];
        int pos = -1;
        if (kk > tau) {
            pos = atomicAdd(&s_hi, 1);
        } else if (kk == tau) {
            int e = atomicAdd(&s_eq, 1);
            if (e < KSEL - n_hi) pos = n_hi + e;
        }
        if (pos >= 0 && pos < KSEL) {
            cv[pos] = key2f(kk);
            ci[pos] = tile * TILE + i;
        }
    }
}

// ---------------------------------------------------------------------------
// Kernel 4: per token merge top-128 of 2048 candidates, coalesce target,
//           masked log-softmax over the sparse set.
// ---------------------------------------------------------------------------
__global__ __launch_bounds__(256) void merge_softmax(const float* __restrict__ cand_val,
                                                     const int* __restrict__ cand_idx,
                                                     const float* __restrict__ tgt_val,
                                                     const int* __restrict__ target,
                                                     int* __restrict__ out_cols,
                                                     float* __restrict__ out_vals,
                                                     int* __restrict__ out_cnt) {
    __shared__ unsigned keys[CAND];               // 8 KB
    __shared__ int s_cnt, s_hi, s_eq, s_found;
    __shared__ int   sel_idx[KSEL];
    __shared__ float sel_val[KSEL];
    __shared__ float red[256];
    const int t = blockIdx.x, tid = threadIdx.x;

    const float* cv = cand_val + (size_t)t * CAND;
    const int*   ci = cand_idx + (size_t)t * CAND;
    for (int i = tid; i < CAND; i += 256) keys[i] = f2key(cv[i]);
    __syncthreads();

    unsigned pivot = 0;
    for (int bit = 31; bit >= 0; --bit) {
        unsigned trial = pivot | (1u << bit);
        if (tid == 0) s_cnt = 0;
        __syncthreads();
        int c = 0;
        for (int i = tid; i < CAND; i += 256) c += (keys[i] >= trial);
        atomicAdd(&s_cnt, c);
        __syncthreads();
        if (s_cnt >= KSEL) pivot = trial;
        __syncthreads();
    }
    const unsigned tau = pivot;

    if (tid == 0) s_hi = 0;
    __syncthreads();
    int cg = 0;
    for (int i = tid; i < CAND; i += 256) cg += (keys[i] > tau);
    atomicAdd(&s_hi, cg);
    __syncthreads();
    const int n_hi = s_hi;
    if (tid == 0) { s_hi = 0; s_eq = 0; s_found = 0; }
    __syncthreads();

    for (int i = tid; i < CAND; i += 256) {
        unsigned kk = keys[i];
        int pos = -1;
        if (kk > tau) {
            pos = atomicAdd(&s_hi, 1);
        } else if (kk == tau) {
            int e = atomicAdd(&s_eq, 1);
            if (e < KSEL - n_hi) pos = n_hi + e;
        }
        if (pos >= 0 && pos < KSEL) {
            sel_val[pos] = key2f(kk);
            sel_idx[pos] = ci[i];
        }
    }
    __syncthreads();

    // Coalesce the target column (sum-on-duplicate semantics).
    const int   tc = target[t];
    const float tv = tgt_val[t];
    for (int i = tid; i < KSEL; i += 256) {
        if (sel_idx[i] == tc) { sel_val[i] += tv; s_found = 1; }
    }
    __syncthreads();
    const int found = s_found;

    // max reduction over the sparse set
    float m = -INFINITY;
    for (int i = tid; i < KSEL; i += 256) m = fmaxf(m, sel_val[i]);
    if (!found && tid == 0) m = fmaxf(m, tv);
    red[tid] = m;
    __syncthreads();
    for (int s = 128; s > 0; s >>= 1) {
        if (tid < s) red[tid] = fmaxf(red[tid], red[tid + s]);
        __syncthreads();
    }
    m = red[0];
    __syncthreads();

    // sum-exp reduction
    float se = 0.0f;
    for (int i = tid; i < KSEL; i += 256) se += expf(sel_val[i] - m);
    if (!found && tid == 0) se += expf(tv - m);
    red[tid] = se;
    __syncthreads();
    for (int s = 128; s > 0; s >>= 1) {
        if (tid < s) red[tid] += red[tid + s];
        __syncthreads();
    }
    const float lse = logf(red[0]);

    int*   oc = out_cols + (size_t)t * ROWPAD;
    float* ov = out_vals + (size_t)t * ROWPAD;
    for (int i = tid; i < KSEL; i += 256) {
        oc[i] = sel_idx[i];
        ov[i] = sel_val[i] - m - lse;
    }
    if (tid == 0) {
        int n = KSEL;
        if (!found) { oc[KSEL] = tc; ov[KSEL] = tv - m - lse; n = KSEL + 1; }
        out_cnt[t] = n;
    }
}

// ---------------------------------------------------------------------------
// Kernel 5: zero the 206 MB output (float4 stores, grid-stride).
// ---------------------------------------------------------------------------
__global__ void zero_out(float4* __restrict__ out, long n4) {
    long i = blockIdx.x * (long)blockDim.x + threadIdx.x;
    const long stride = (long)gridDim.x * blockDim.x;
    const float4 z = make_float4(0.f, 0.f, 0.f, 0.f);
    for (; i < n4; i += stride) out[i] = z;
}

// ---------------------------------------------------------------------------
// Kernel 6: scatter sparse log-softmax entries.
// ---------------------------------------------------------------------------
__global__ void scatter(const int* __restrict__ out_cols,
                        const float* __restrict__ out_vals,
                        const int* __restrict__ out_cnt,
                        float* __restrict__ out) {
    const int t = blockIdx.x;
    const int n = out_cnt[t];
    for (int i = threadIdx.x; i < n; i += blockDim.x)
        out[(size_t)t * VOCAB + out_cols[(size_t)t * ROWPAD + i]] =
            out_vals[(size_t)t * ROWPAD + i];
}

// ---------------------------------------------------------------------------
extern "C" void kernel_launch(void* const* d_in, const int* in_sizes, int n_in,
                              void* d_out, int out_size, void* d_ws, size_t ws_size,
                              hipStream_t stream) {
    const float* x    = (const float*)d_in[0];
    const float* w    = (const float*)d_in[1];
    const int*   tgt  = (const int*)d_in[2];
    float* out = (float*)d_out;

    // Workspace layout (bytes).
    char* ws = (char*)d_ws;
    float* cand_val = (float*)(ws);                                   // 8 MB
    int*   cand_idx = (int*)(ws + (size_t)T_TOK * CAND * 4);          // 8 MB
    float* tgt_val  = (float*)(ws + (size_t)T_TOK * CAND * 8);        // 4 KB
    int*   out_cols = (int*)(ws + (size_t)T_TOK * CAND * 8 + T_TOK * 4);
    float* out_vals = (float*)((char*)out_cols + (size_t)T_TOK * ROWPAD * 4);
    int*   out_cnt  = (int*)((char*)out_vals + (size_t)T_TOK * ROWPAD * 4);

    // 1) GEMM: logits staged directly into d_out.
    dim3 gA(TILE * TILEF / 64, T_TOK / MROWS);    // (786, 16)
    gemm_logits<<<gA, 128, 0, stream>>>(x, w, out);

    // 2) Target logits (read before d_out is recycled).
    grab_target<<<(T_TOK + 255) / 256, 256, 0, stream>>>(out, tgt, tgt_val);

    // 3) Per-tile top-128.
    dim3 gB(TILEF, T_TOK);
    tile_topk<<<gB, 256, 0, stream>>>(out, cand_val, cand_idx);

    // 4) Merge + coalesce target + sparse log-softmax.
    merge_softmax<<<T_TOK, 256, 0, stream>>>(cand_val, cand_idx, tgt_val, tgt,
                                             out_cols, out_vals, out_cnt);

    // 5) Zero the dense output, 6) scatter sparse entries.
    const long n4 = (long)T_TOK * VOCAB / 4;
    zero_out<<<2048, 256, 0, stream>>>((float4*)out, n4);
    scatter<<<T_TOK, 256, 0, stream>>>(out_cols, out_vals, out_cnt, out);
}